// ProdLayer_49813030699584
// MI455X (gfx1250) — compile-verified
//
#include <hip/hip_runtime.h>

#define GROUP_SIZE 16
#define N_EDGES    8
#define BATCH      64
#define TPB        256   // 8 wave32; thread t owns byte-offset 16*t of every 4KB tile

typedef float v4f __attribute__((ext_vector_type(4)));
typedef int   v4i __attribute__((ext_vector_type(4)));

typedef __attribute__((address_space(1))) v4i* gptr_b128;   // global, 16B granule
typedef __attribute__((address_space(3))) v4i* lptr_b128;   // LDS, 16B granule

#define HAS_ASYNC_LDS (__has_builtin(__builtin_amdgcn_global_load_async_to_lds_b128) && \
                       __has_builtin(__builtin_amdgcn_s_wait_asynccnt))

#if HAS_ASYNC_LDS
// ---------------------------------------------------------------------------
// CDNA5 async path: stage the eight 16x64 f32 edge tiles (4 KB each) into LDS
// with GLOBAL_LOAD_ASYNC_TO_LDS_B128, wait on ASYNCcnt, reduce from LDS.
// Thread t writes and reads only its own 16 B slot per edge -> no barrier,
// only the per-wave s_wait_asynccnt.
// ---------------------------------------------------------------------------
__global__ __launch_bounds__(TPB)
void prod_layer_gather_sum(const float* __restrict__ node_mars,
                           const int*   __restrict__ nids,
                           const int*   __restrict__ cids,
                           float*       __restrict__ out)
{
    __shared__ v4f tiles[N_EDGES * TPB];          // 8 * 4 KB = 32 KB LDS

    const int n = blockIdx.x;
    const int t = threadIdx.x;

    const int nid  = nids[n];                     // scalar load
    const int* crow = cids + (size_t)n * N_EDGES; // s_load_b256

#pragma unroll
    for (int e = 0; e < N_EDGES; ++e) {
        // Uniform tile base (SGPR) + shared 16*t byte lane offset.
        const v4f* src = reinterpret_cast<const v4f*>(
                             node_mars + ((size_t)crow[e] << 6)) + t;
        __builtin_amdgcn_global_load_async_to_lds_b128(
            (gptr_b128)(void*)(const void*)src,
            (lptr_b128)(void*)&tiles[e * TPB + t],
            /*imm offset*/0, /*cpol*/0);
    }

    __builtin_amdgcn_s_wait_asynccnt(0);

    v4f acc = (v4f){0.f, 0.f, 0.f, 0.f};
#pragma unroll
    for (int e = 0; e < N_EDGES; ++e)
        acc += tiles[e * TPB + t];

    // Output written once, never re-read: non-temporal store (TH=NT).
    v4f* o = reinterpret_cast<v4f*>(out + ((size_t)nid << 10)) + t;
    __builtin_nontemporal_store(acc, o);
}
#else
// ---------------------------------------------------------------------------
// Fallback: direct global_load_b128 gathers, GVS addressing
// (scalar tile base + shared 32-bit lane offset 16*t).
// ---------------------------------------------------------------------------
__global__ __launch_bounds__(TPB)
void prod_layer_gather_sum(const float* __restrict__ node_mars,
                           const int*   __restrict__ nids,
                           const int*   __restrict__ cids,
                           float*       __restrict__ out)
{
    const int n = blockIdx.x;
    const int t = threadIdx.x;

    const int nid  = nids[n];
    const int* crow = cids + (size_t)n * N_EDGES;

    v4f acc = (v4f){0.f, 0.f, 0.f, 0.f};
#pragma unroll
    for (int e = 0; e < N_EDGES; ++e) {
        const v4f* src = reinterpret_cast<const v4f*>(
                             node_mars + ((size_t)crow[e] << 6)) + t;
        acc += *src;   // temporal: node rows are revisited across groups, L2=192MB
    }

    v4f* o = reinterpret_cast<v4f*>(out + ((size_t)nid << 10)) + t;
    __builtin_nontemporal_store(acc, o);
}
#endif

extern "C" void kernel_launch(void* const* d_in, const int* in_sizes, int n_in,
                              void* d_out, int out_size, void* d_ws, size_t ws_size,
                              hipStream_t stream)
{
    const float* node_mars = (const float*)d_in[0];
    const int*   nids      = (const int*)d_in[2];
    const int*   cids      = (const int*)d_in[3];
    float*       out       = (float*)d_out;

    const int n_groups = in_sizes[2];

    // General-case guard: seed d_out with element_mars if the written rows
    // would not cover it. With these shapes (32768*16*64 == out_size) the
    // copy is skipped.
    const long long covered = (long long)n_groups * GROUP_SIZE * BATCH;
    if (covered < (long long)out_size) {
        (void)hipMemcpyAsync(d_out, d_in[1], (size_t)out_size * sizeof(float),
                             hipMemcpyDeviceToDevice, stream);
    }

    prod_layer_gather_sum<<<dim3(n_groups), dim3(TPB), 0, stream>>>(
        node_mars, nids, cids, out);
}